// PostProcessor_1889785610762
// MI455X (gfx1250) — compile-verified
//
#include <hip/hip_runtime.h>
#include <math.h>

// ---------------- problem constants (from setup_inputs) ----------------
#define BN   16
#define CN   10
#define HN   128
#define WN   352
#define HWp  (HN*WN)        // 45056
#define TK   100
#define RCH  50
#define NPAIR 2048
#define NBINS 1024
#define CAND  2048

// NMS tiling
#define TH_  16
#define TW_  32
#define TILEH (TH_+2)
#define TILEW (TW_+2)
#define TILEN (TILEH*TILEW)          // 612
#define TPX (WN/TW_)                 // 11
#define TPY (HN/TH_)                 // 8
#define BLOCKS_PLANE (TPX*TPY)       // 88
#define NMS_BLOCKS (BN*CN*BLOCKS_PLANE)

// output layout (flat concat in reference return order)
#define O_SC   0
#define O_CL   (BN*TK)
#define O_XS   (2*BN*TK)
#define O_YS   (3*BN*TK)
#define O_POIS (4*BN*TK)
#define O_MASK (O_POIS + BN*TK*RCH)  // 86400
#define O_IOU  (O_MASK + BN*TK)      // 88000

// ---------------- CDNA5 feature probes ----------------
#if defined(__has_builtin)
#  if __has_builtin(__builtin_amdgcn_global_load_async_to_lds_b32)
#    define HAS_ASYNC_LDS 1
#  endif
#  if __has_builtin(__builtin_amdgcn_s_wait_asynccnt)
#    define HAS_WAIT_ASYNC 1
#  endif
#  if __has_builtin(__builtin_amdgcn_wmma_f32_16x16x4_f32)
#    define HAS_WMMA_F32X4 1
#  endif
#endif

typedef __attribute__((address_space(1))) int* gint_p;
typedef __attribute__((address_space(3))) int* lint_p;

__device__ __forceinline__ void async_g2l_b32(const float* g, float* l) {
#ifdef HAS_ASYNC_LDS
  float* gm = const_cast<float*>(g);
  __builtin_amdgcn_global_load_async_to_lds_b32((gint_p)gm, (lint_p)l, 0, 0);
#else
  *l = *g;
#endif
}

__device__ __forceinline__ void wait_async_lds() {
#if defined(HAS_WAIT_ASYNC)
  __builtin_amdgcn_s_wait_asynccnt(0);
#elif defined(HAS_ASYNC_LDS)
  asm volatile("s_wait_asynccnt 0x0" ::: "memory");
#endif
}

// ---------------- kernel 0: zero workspace counters ----------------
__global__ void k_init(int* __restrict__ ghist, int* __restrict__ cnt) {
  int i = blockIdx.x * blockDim.x + threadIdx.x;
  if (i < BN * NBINS) ghist[i] = 0;
  if (i < BN) cnt[i] = 0;
}

// ---------------- kernel 1/3: fused 3x3-NMS (+ histogram | + collect) ----
// MODE 0: build per-batch value histogram of the NMS-ed heatmap.
// MODE 1: collect (value, flat CHW index) candidates with bin >= thr[b].
template <int MODE>
__global__ __launch_bounds__(256) void k_nms(const float* __restrict__ hm,
                                             int* __restrict__ ghist,
                                             const int* __restrict__ thr,
                                             int* __restrict__ cnt,
                                             float* __restrict__ cval,
                                             int* __restrict__ cidx) {
  __shared__ float tile[TILEN];
  __shared__ int hist[NBINS];

  const int tid = threadIdx.x;
  const int blk = blockIdx.x;
  const int plane = blk / BLOCKS_PLANE;       // b*CN + c
  const int tb = blk % BLOCKS_PLANE;
  const int b = plane / CN;
  const int c = plane % CN;
  const int y0 = (tb / TPX) * TH_;
  const int x0 = (tb % TPX) * TW_;
  const float* base = hm + (size_t)plane * HWp;

  if (MODE == 0)
    for (int i = tid; i < NBINS; i += 256) hist[i] = 0;
  for (int i = tid; i < TILEN; i += 256) tile[i] = -__builtin_inff();
  __syncthreads();

  // async-copy the halo tile from HBM into LDS (out-of-plane stays -inf)
  for (int i = tid; i < TILEN; i += 256) {
    int ty = i / TILEW, tx = i % TILEW;
    int gy = y0 - 1 + ty, gx = x0 - 1 + tx;
    if (gy >= 0 && gy < HN && gx >= 0 && gx < WN)
      async_g2l_b32(base + (size_t)gy * WN + gx, &tile[i]);
  }
  wait_async_lds();
  __syncthreads();

  for (int e = tid; e < TH_ * TW_; e += 256) {
    int ty = e / TW_, tx = e % TW_;
    float v = tile[(ty + 1) * TILEW + (tx + 1)];
    float m = v;
#pragma unroll
    for (int dy = 0; dy < 3; ++dy)
#pragma unroll
      for (int dx = 0; dx < 3; ++dx)
        m = fmaxf(m, tile[(ty + dy) * TILEW + (tx + dx)]);
    float val = (v == m) ? v : 0.0f;      // hm * (pooled == hm)
    int bin = (int)(val * (float)NBINS);
    bin = bin < 0 ? 0 : (bin > NBINS - 1 ? NBINS - 1 : bin);
    if (MODE == 0) {
      atomicAdd(&hist[bin], 1);
    } else {
      if (bin >= thr[b]) {
        int pos = atomicAdd(&cnt[b], 1);
        if (pos < CAND) {
          cval[b * CAND + pos] = val;
          cidx[b * CAND + pos] = c * HWp + (y0 + ty) * WN + (x0 + tx);
        }
      }
    }
  }

  if (MODE == 0) {
    __syncthreads();
    for (int i = tid; i < NBINS; i += 256) {
      int h = hist[i];
      if (h) atomicAdd(&ghist[b * NBINS + i], h);
    }
  }
}

// ---------------- kernel 2: per-batch threshold bin ----------------
__global__ void k_select(const int* __restrict__ ghist, int* __restrict__ thr,
                         const int* __restrict__ Kp) {
  int b = threadIdx.x;
  if (b >= BN) return;
  int K = Kp[0];
  int acc = 0, t = 0;
  for (int i = NBINS - 1; i >= 0; --i) {
    acc += ghist[b * NBINS + i];
    if (acc >= K) { t = i; break; }
  }
  thr[b] = t;
}

// ---------------- kernel 4: exact top-K + outputs + POI gather --------
__global__ __launch_bounds__(256) void k_final(const float* __restrict__ reg,
                                               const float* __restrict__ cval,
                                               const int* __restrict__ cidx,
                                               const int* __restrict__ cnt,
                                               float* __restrict__ out) {
  __shared__ float lv[CAND];
  __shared__ int li[CAND];
  __shared__ float sv[256];
  __shared__ int sid[256];
  __shared__ int spos[256];
  __shared__ int selSpat[TK];

  const int b = blockIdx.x, tid = threadIdx.x;
  int nn = cnt[b];
  nn = nn < CAND ? nn : CAND;
  for (int j = tid; j < nn; j += 256) {
    lv[j] = cval[b * CAND + j];
    li[j] = cidx[b * CAND + j];
  }
  __syncthreads();

  for (int k = 0; k < TK; ++k) {
    float bv = -2.0f; int bid = 0x7fffffff; int bp = -1;
    for (int j = tid; j < nn; j += 256) {
      float v = lv[j]; int id = li[j];
      if (v > bv || (v == bv && id < bid)) { bv = v; bid = id; bp = j; }
    }
    sv[tid] = bv; sid[tid] = bid; spos[tid] = bp;
    __syncthreads();
    for (int s = 128; s > 0; s >>= 1) {
      if (tid < s) {
        float v2 = sv[tid + s]; int id2 = sid[tid + s];
        if (v2 > sv[tid] || (v2 == sv[tid] && id2 < sid[tid])) {
          sv[tid] = v2; sid[tid] = id2; spos[tid] = spos[tid + s];
        }
      }
      __syncthreads();
    }
    if (tid == 0) {
      float v = sv[0]; int id = sid[0]; int pos = spos[0];
      if (pos >= 0) lv[pos] = -1.0f; else { v = 0.0f; id = 0; }
      int cls = id / HWp;
      int sp = id % HWp;
      out[O_SC + b * TK + k] = v;
      out[O_CL + b * TK + k] = (float)cls;
      out[O_XS + b * TK + k] = (float)(sp % WN);
      out[O_YS + b * TK + k] = (float)(sp / WN);
      out[O_MASK + b * TK + k] = (v >= 0.25f) ? 1.0f : 0.0f;
      selSpat[k] = sp;
    }
    __syncthreads();
  }

  // pois[b][k][r] = reg[b][r][spatial]
  for (int t = tid; t < TK * RCH; t += 256) {
    int k = t / RCH, r = t % RCH;
    out[O_POIS + (size_t)(b * TK + k) * RCH + r] =
        reg[((size_t)b * RCH + r) * HWp + selSpat[k]];
  }
}

// ---------------- kernel 5: rotated 3D IoU ----------------
typedef __attribute__((ext_vector_type(2))) float v2f;
typedef __attribute__((ext_vector_type(8))) float v8f;

__device__ __forceinline__ float shoelace4(const float* x, const float* y) {
  float s = 0.f;
#pragma unroll
  for (int i = 0; i < 4; ++i) { int i2 = (i + 1) & 3; s += x[i] * y[i2] - y[i] * x[i2]; }
  return 0.5f * fabsf(s);
}

__device__ float quad_inter_area(const float* ax, const float* az,
                                 const float* bx, const float* bz) {
  const float EPSF = 1e-8f;
  float px[24], py[24];
  int vld[24];
  // 16 edge-pair intersections
  for (int i = 0; i < 4; ++i) {
    int i2 = (i + 1) & 3;
    float d1x = ax[i2] - ax[i], d1y = az[i2] - az[i];
    for (int j = 0; j < 4; ++j) {
      int j2 = (j + 1) & 3;
      float d2x = bx[j2] - bx[j], d2y = bz[j2] - bz[j];
      float den = d1x * d2y - d1y * d2x;
      float safe = (fabsf(den) < EPSF) ? 1.0f : den;
      float dqx = bx[j] - ax[i], dqy = bz[j] - az[i];
      float t = (dqx * d2y - dqy * d2x) / safe;
      float u = (dqx * d1y - dqy * d1x) / safe;
      int k = i * 4 + j;
      px[k] = ax[i] + t * d1x;
      py[k] = az[i] + t * d1y;
      vld[k] = (fabsf(den) >= EPSF) && (t >= 0.f) && (t <= 1.f) && (u >= 0.f) && (u <= 1.f);
    }
  }
  // A corners inside B, B corners inside A
  for (int q = 0; q < 2; ++q) {
    const float* qx = q ? ax : bx; const float* qz = q ? az : bz; // quad
    const float* ppx = q ? bx : ax; const float* ppz = q ? bz : az; // points
    for (int i = 0; i < 4; ++i) {
      bool allp = true, alln = true;
      for (int e = 0; e < 4; ++e) {
        int e2 = (e + 1) & 3;
        float cr = (qx[e2] - qx[e]) * (ppz[i] - qz[e]) - (qz[e2] - qz[e]) * (ppx[i] - qx[e]);
        allp = allp && (cr >= -1e-9f);
        alln = alln && (cr <= 1e-9f);
      }
      int k = 16 + (1 - q) * 4 + i;  // q=1 -> B-in-A at 16..19 ; q=0 -> A-in-B at 20..23
      px[k] = ppx[i]; py[k] = ppz[i]; vld[k] = (allp || alln);
    }
  }
  int nv = 0; float cxs = 0.f, cys = 0.f;
  for (int i = 0; i < 24; ++i) if (vld[i]) { ++nv; cxs += px[i]; cys += py[i]; }
  if (nv < 3) return 0.f;
  float cx = cxs / (float)nv, cy = cys / (float)nv;
  float ang[24];
  for (int i = 0; i < 24; ++i)
    ang[i] = vld[i] ? atan2f(py[i] - cy, px[i] - cx) : __builtin_inff();
  // stable insertion sort by angle
  for (int i = 1; i < 24; ++i) {
    float a = ang[i], xx = px[i], yy = py[i]; int vv = vld[i];
    int j = i - 1;
    while (j >= 0 && ang[j] > a) {
      ang[j + 1] = ang[j]; px[j + 1] = px[j]; py[j + 1] = py[j]; vld[j + 1] = vld[j]; --j;
    }
    ang[j + 1] = a; px[j + 1] = xx; py[j + 1] = yy; vld[j + 1] = vv;
  }
  float fx = px[0], fy = py[0];
  for (int i = 0; i < 24; ++i) if (!vld[i]) { px[i] = fx; py[i] = fy; }
  float s = 0.f;
  for (int i = 0; i < 24; ++i) { int i2 = (i + 1) % 24; s += px[i] * py[i2] - py[i] * px[i2]; }
  return 0.5f * fabsf(s);
}

// One wave (32 lanes) per block: WMMA-batched corner GEMM for 32 box pairs,
// corners staged through LDS, then one lane per pair does the clipping.
__global__ __launch_bounds__(32) void k_iou(const float* __restrict__ pred,
                                            const float* __restrict__ tgt,
                                            float* __restrict__ iou_out) {
  __shared__ float corn[32][16];  // [pair][pred cx0..3 cz0..3 | tgt cx0..3 cz0..3]
  const int lane = threadIdx.x;
  const int base = blockIdx.x * 32;
  const int n = lane & 15;
  const bool hi = lane >= 16;
  const float sxl[4] = {0.5f, 0.5f, -0.5f, -0.5f};
  const float szl[4] = {0.5f, -0.5f, -0.5f, 0.5f};

#ifdef HAS_WMMA_F32X4
  // B matrices (4x16 f32): cols 0..3 = cx_j, cols 4..7 = cz_j, cols 8..15 = 0.
  // B1 rows = features {x, z, c*l, s*w}; B2 rows = {s*l, c*w, 0, 0}.
  const int j4 = n & 3;
  const bool isCz = (n >= 4) && (n < 8);
  const bool act = n < 8;
  v2f b1 = {0.f, 0.f}, b2 = {0.f, 0.f};
  if (act) {
    if (!hi) {  // B rows 0,1
      b1.x = isCz ? 0.f : 1.f;           // x  -> cx
      b1.y = isCz ? 1.f : 0.f;           // z  -> cz
      b2.x = isCz ? sxl[j4] : 0.f;       // s*l -> cz
      b2.y = isCz ? szl[j4] : 0.f;       // c*w -> cz
    } else {    // B rows 2,3
      b1.x = isCz ? 0.f : sxl[j4];       // c*l -> cx
      b1.y = isCz ? 0.f : -szl[j4];      // s*w -> cx
      // b2 rows 2,3 are zero padding
    }
  }
#pragma unroll
  for (int q = 0; q < 4; ++q) {          // {pred,pred,tgt,tgt} x {lo16,hi16}
    const float* src = (q < 2) ? pred : tgt;
    const int bi = base + (q & 1) * 16 + n;
    float x = src[bi * 7 + 0], z = src[bi * 7 + 2];
    float l = src[bi * 7 + 3], w = src[bi * 7 + 5];
    float ry = src[bi * 7 + 6];
    float sn, cs;
    sincosf(ry, &sn, &cs);
    v2f a1, a2;  // 16x4 f32 A: lanes0-15 = K0,K1 ; lanes16-31 = K2,K3
    a1.x = hi ? cs * l : x;
    a1.y = hi ? sn * w : z;
    a2.x = hi ? 0.f : sn * l;
    a2.y = hi ? 0.f : cs * w;
    v8f d = {0.f, 0.f, 0.f, 0.f, 0.f, 0.f, 0.f, 0.f};
    d = __builtin_amdgcn_wmma_f32_16x16x4_f32(false, a1, false, b1, (short)0, d, false, false);
    d = __builtin_amdgcn_wmma_f32_16x16x4_f32(false, a2, false, b2, (short)0, d, false, false);
    if (n < 8) {
      const int coff = (q < 2) ? 0 : 8;
#pragma unroll
      for (int r = 0; r < 8; ++r) {
        int pr = (q & 1) * 16 + r + (hi ? 8 : 0);  // pair-in-block (row M)
        corn[pr][coff + n] = d[r];
      }
    }
  }
#else
  // scalar fallback: each lane computes its own pair's corners
  {
    const int p = base + lane;
    for (int q = 0; q < 2; ++q) {
      const float* src = q ? tgt : pred;
      float x = src[p * 7 + 0], z = src[p * 7 + 2];
      float l = src[p * 7 + 3], w = src[p * 7 + 5];
      float ry = src[p * 7 + 6];
      float sn, cs;
      sincosf(ry, &sn, &cs);
      for (int jj = 0; jj < 4; ++jj) {
        float dx = sxl[jj] * l, dz = szl[jj] * w;
        corn[lane][q * 8 + jj] = x + cs * dx - sn * dz;
        corn[lane][q * 8 + 4 + jj] = z + sn * dx + cs * dz;
      }
    }
  }
#endif
  __syncthreads();

  const int p = base + lane;
  float axp[4], azp[4], bxp[4], bzp[4];
#pragma unroll
  for (int i = 0; i < 4; ++i) {
    axp[i] = corn[lane][i];     azp[i] = corn[lane][4 + i];
    bxp[i] = corn[lane][8 + i]; bzp[i] = corn[lane][12 + i];
  }
  float p1 = pred[p * 7 + 1], p4 = pred[p * 7 + 4];
  float t1 = tgt[p * 7 + 1],  t4 = tgt[p * 7 + 4];
  float minA = -(p1 + 0.5f * p4), maxA = -(p1 - 0.5f * p4);
  float minB = -(t1 + 0.5f * t4), maxB = -(t1 - 0.5f * t4);
  float hov = fminf(maxA, maxB) - fmaxf(minA, minB);
  hov = hov > 0.f ? hov : 0.f;
  float areaA = shoelace4(axp, azp);
  float areaB = shoelace4(bxp, bzp);
  float inter = quad_inter_area(axp, azp, bxp, bzp) * hov;
  float h0 = pred[0 * 7 + 4];  // reference uses max_h_a[0]-min_h_a[0] == pred[0,4]
  float uni = areaA * h0 + areaB * t4 - inter;
  iou_out[p] = inter / uni;
}

// ---------------- launcher ----------------
extern "C" void kernel_launch(void* const* d_in, const int* in_sizes, int n_in,
                              void* d_out, int out_size, void* d_ws, size_t ws_size,
                              hipStream_t stream) {
  const float* hm  = (const float*)d_in[0];   // (16,10,128,352)
  const float* reg = (const float*)d_in[1];   // (16,50,128,352)
  const float* pb  = (const float*)d_in[2];   // (2048,7)
  const float* tb  = (const float*)d_in[3];   // (2048,7)
  const int*   Kp  = (const int*)d_in[4];     // scalar K=100
  float* out = (float*)d_out;

  char* ws = (char*)d_ws;
  int*   ghist = (int*)ws;                                   // 16*1024*4 = 65536 B
  int*   thr   = (int*)(ws + 65536);                         // 64 B
  int*   cnt   = (int*)(ws + 65536 + 64);                    // 64 B
  float* cval  = (float*)(ws + 65536 + 128);                 // 16*2048*4 = 131072 B
  int*   cidx  = (int*)(ws + 65536 + 128 + 131072);          // 131072 B

  k_init<<<(BN * NBINS + 255) / 256, 256, 0, stream>>>(ghist, cnt);
  k_nms<0><<<NMS_BLOCKS, 256, 0, stream>>>(hm, ghist, nullptr, nullptr, nullptr, nullptr);
  k_select<<<1, 32, 0, stream>>>(ghist, thr, Kp);
  k_nms<1><<<NMS_BLOCKS, 256, 0, stream>>>(hm, nullptr, thr, cnt, cval, cidx);
  k_final<<<BN, 256, 0, stream>>>(reg, cval, cidx, cnt, out);
  k_iou<<<NPAIR / 32, 32, 0, stream>>>(pb, tb, out + O_IOU);
}